// VoxelModule_68393059221508
// MI455X (gfx1250) — compile-verified
//
#include <hip/hip_runtime.h>
#include <hip/hip_bf16.h>
#include <stdint.h>

// ---------------- problem constants (fixed by the reference harness) --------
constexpr int B = 32;
constexpr int N = 262144;                       // points per batch
constexpr int BATCH_FLOATS = N * 3;             // 786432
constexpr int BATCH_BYTES  = BATCH_FLOATS * 4;  // 3,145,728

// reduction kernel geometry
constexpr int R_BLOCKS_PER_BATCH = 32;
constexpr int R_WAVES_PER_BLOCK  = 8;           // 256 threads, wave32
constexpr int CHUNK_BYTES        = 32 * 48;     // 32 lanes x 4 points x 12B = 1536
constexpr int WAVE_SLICE_BYTES   = BATCH_BYTES / (R_BLOCKS_PER_BATCH * R_WAVES_PER_BLOCK); // 12288
constexpr int R_ITERS            = WAVE_SLICE_BYTES / CHUNK_BYTES;   // 8
static_assert(WAVE_SLICE_BYTES * R_BLOCKS_PER_BATCH * R_WAVES_PER_BLOCK == BATCH_BYTES, "slice");
static_assert(R_ITERS * CHUNK_BYTES == WAVE_SLICE_BYTES, "iters");

// assign kernel geometry: 256 threads x 4 points = 1024 points per block
constexpr int A_BLOCKS_PER_BATCH = N / 1024;    // 256
static_assert(A_BLOCKS_PER_BATCH * 1024 == N, "assign");

// ws layout: ws[0..95] = per-(batch,coord) min, ws[96..191] = max
constexpr int WS_MAX_OFF_BYTES = 96 * 4;

typedef __attribute__((ext_vector_type(4))) float v4f;

// ---------------------------------------------------------------------------
__global__ void voxel_init_kernel(float* __restrict__ ws) {
    const int i = threadIdx.x;            // 96 threads
    ws[i]      =  __builtin_inff();
    ws[96 + i] = -__builtin_inff();
}

__device__ __forceinline__ float wave_min32(float v) {
    #pragma unroll
    for (int m = 16; m >= 1; m >>= 1) v = fminf(v, __shfl_xor(v, m, 32));
    return v;
}
__device__ __forceinline__ float wave_max32(float v) {
    #pragma unroll
    for (int m = 16; m >= 1; m >>= 1) v = fmaxf(v, __shfl_xor(v, m, 32));
    return v;
}

// Pass 1: per-(batch,coord) min/max via async global->LDS streaming + float atomics.
// Default RT temporal hints deliberately leave the 100.7 MB input resident in the
// 192 MB L2 so pass 2 reads hit L2 instead of HBM.
__global__ __launch_bounds__(256) void voxel_minmax_kernel(const float* __restrict__ pc,
                                                           float* __restrict__ ws) {
    __shared__ __align__(16) unsigned char smem[R_WAVES_PER_BLOCK * 2 * CHUNK_BYTES]; // 24 KB
    __shared__ float wmin[R_WAVES_PER_BLOCK][3];
    __shared__ float wmax[R_WAVES_PER_BLOCK][3];

    const int batch = blockIdx.x / R_BLOCKS_PER_BATCH;
    const int blk   = blockIdx.x % R_BLOCKS_PER_BATCH;
    const int wave  = threadIdx.x >> 5;
    const int lane  = threadIdx.x & 31;

    // block-uniform global base (lands in an SGPR pair for the GVS async form)
    const char* gbase = (const char*)pc + (size_t)batch * (size_t)BATCH_BYTES;

    const uint32_t waveBase = (uint32_t)(blk * R_WAVES_PER_BLOCK + wave) * (uint32_t)WAVE_SLICE_BYTES;
    const uint32_t ldsWave  = (uint32_t)(uintptr_t)(&smem[0]) + (uint32_t)wave * (2u * CHUNK_BYTES);
    const uint32_t lane48   = (uint32_t)lane * 48u;

    // prime buffer 0 with chunk 0 (each lane deposits 48B = 4 points, 3 x b128)
    {
        uint32_t ldsa = ldsWave + lane48;
        uint32_t goff = waveBase + lane48;
        asm volatile(
            "global_load_async_to_lds_b128 %0, %1, %2 offset:0\n\t"
            "global_load_async_to_lds_b128 %0, %1, %2 offset:16\n\t"
            "global_load_async_to_lds_b128 %0, %1, %2 offset:32"
            :: "v"(ldsa), "v"(goff), "s"(gbase) : "memory");
    }

    float mn0 =  __builtin_inff(), mn1 =  __builtin_inff(), mn2 =  __builtin_inff();
    float mx0 = -__builtin_inff(), mx1 = -__builtin_inff(), mx2 = -__builtin_inff();

    for (int it = 0; it < R_ITERS; ++it) {
        const int buf = it & 1;
        if (it + 1 < R_ITERS) {
            // reuse hazard: prior ds_reads of the buffer being overwritten must finish
            asm volatile("s_wait_dscnt 0" ::: "memory");
            uint32_t ldsa = ldsWave + (uint32_t)((it + 1) & 1) * (uint32_t)CHUNK_BYTES + lane48;
            uint32_t goff = waveBase + (uint32_t)(it + 1) * (uint32_t)CHUNK_BYTES + lane48;
            asm volatile(
                "global_load_async_to_lds_b128 %0, %1, %2 offset:0\n\t"
                "global_load_async_to_lds_b128 %0, %1, %2 offset:16\n\t"
                "global_load_async_to_lds_b128 %0, %1, %2 offset:32"
                :: "v"(ldsa), "v"(goff), "s"(gbase) : "memory");
            // 6 outstanding, in-order completion: <=3 pending => current chunk landed
            asm volatile("s_wait_asynccnt 3" ::: "memory");
        } else {
            asm volatile("s_wait_asynccnt 0" ::: "memory");
        }

        const float4* q = (const float4*)(&smem[0] + wave * (2 * CHUNK_BYTES)
                                          + buf * CHUNK_BYTES + lane * 48);
        float4 a = q[0];  // x0 y0 z0 x1
        float4 b = q[1];  // y1 z1 x2 y2
        float4 c = q[2];  // z2 x3 y3 z3

        mn0 = fminf(mn0, fminf(fminf(a.x, a.w), fminf(b.z, c.y)));
        mx0 = fmaxf(mx0, fmaxf(fmaxf(a.x, a.w), fmaxf(b.z, c.y)));
        mn1 = fminf(mn1, fminf(fminf(a.y, b.x), fminf(b.w, c.z)));
        mx1 = fmaxf(mx1, fmaxf(fmaxf(a.y, b.x), fmaxf(b.w, c.z)));
        mn2 = fminf(mn2, fminf(fminf(a.z, b.y), fminf(c.x, c.w)));
        mx2 = fmaxf(mx2, fmaxf(fmaxf(a.z, b.y), fmaxf(c.x, c.w)));
    }

    mn0 = wave_min32(mn0); mn1 = wave_min32(mn1); mn2 = wave_min32(mn2);
    mx0 = wave_max32(mx0); mx1 = wave_max32(mx1); mx2 = wave_max32(mx2);

    if (lane == 0) {
        wmin[wave][0] = mn0; wmin[wave][1] = mn1; wmin[wave][2] = mn2;
        wmax[wave][0] = mx0; wmax[wave][1] = mx1; wmax[wave][2] = mx2;
    }
    __syncthreads();

    if (threadIdx.x == 0) {
        #pragma unroll
        for (int c = 0; c < 3; ++c) {
            float mn = wmin[0][c], mx = wmax[0][c];
            #pragma unroll
            for (int w = 1; w < R_WAVES_PER_BLOCK; ++w) {
                mn = fminf(mn, wmin[w][c]);
                mx = fmaxf(mx, wmax[w][c]);
            }
            uint32_t offMn = (uint32_t)(batch * 3 + c) * 4u;
            uint32_t offMx = offMn + (uint32_t)WS_MAX_OFF_BYTES;
            asm volatile("global_atomic_min_num_f32 %0, %1, %2 scope:SCOPE_DEV"
                         :: "v"(offMn), "v"(mn), "s"(ws) : "memory");
            asm volatile("global_atomic_max_num_f32 %0, %1, %2 scope:SCOPE_DEV"
                         :: "v"(offMx), "v"(mx), "s"(ws) : "memory");
        }
    }
}

// Pass 2: voxels = floor((p - min) / bin), bin = (max - min) / 40.
// Input loads are NT (last use of L2-resident lines); output stores are NT so the
// never-re-read 100 MB result doesn't evict input lines other blocks still need.
__global__ __launch_bounds__(256) void voxel_assign_kernel(const float* __restrict__ pc,
                                                           const float* __restrict__ ws,
                                                           float* __restrict__ out) {
    const int batch = blockIdx.x >> 8;            // 256 blocks per batch
    const int blk   = blockIdx.x & 255;
    const int tid   = threadIdx.x;

    const float mn0 = ws[batch * 3 + 0];
    const float mn1 = ws[batch * 3 + 1];
    const float mn2 = ws[batch * 3 + 2];
    const float bw0 = (ws[96 + batch * 3 + 0] - mn0) / 40.0f;
    const float bw1 = (ws[96 + batch * 3 + 1] - mn1) / 40.0f;
    const float bw2 = (ws[96 + batch * 3 + 2] - mn2) / 40.0f;
    const float iv0 = 1.0f / bw0;
    const float iv1 = 1.0f / bw1;
    const float iv2 = 1.0f / bw2;

    const size_t fbase = ((size_t)batch * N + (size_t)blk * 1024 + (size_t)tid * 4) * 3;
    const v4f* in4 = (const v4f*)(pc + fbase);
    v4f a = __builtin_nontemporal_load(in4 + 0);   // x0 y0 z0 x1
    v4f b = __builtin_nontemporal_load(in4 + 1);   // y1 z1 x2 y2
    v4f c = __builtin_nontemporal_load(in4 + 2);   // z2 x3 y3 z3

    a.x = floorf((a.x - mn0) * iv0);
    a.y = floorf((a.y - mn1) * iv1);
    a.z = floorf((a.z - mn2) * iv2);
    a.w = floorf((a.w - mn0) * iv0);

    b.x = floorf((b.x - mn1) * iv1);
    b.y = floorf((b.y - mn2) * iv2);
    b.z = floorf((b.z - mn0) * iv0);
    b.w = floorf((b.w - mn1) * iv1);

    c.x = floorf((c.x - mn2) * iv2);
    c.y = floorf((c.y - mn0) * iv0);
    c.z = floorf((c.z - mn1) * iv1);
    c.w = floorf((c.w - mn2) * iv2);

    v4f* o4 = (v4f*)(out + fbase);
    __builtin_nontemporal_store(a, o4 + 0);
    __builtin_nontemporal_store(b, o4 + 1);
    __builtin_nontemporal_store(c, o4 + 2);
}

extern "C" void kernel_launch(void* const* d_in, const int* in_sizes, int n_in,
                              void* d_out, int out_size, void* d_ws, size_t ws_size,
                              hipStream_t stream) {
    (void)in_sizes; (void)n_in; (void)out_size; (void)ws_size;
    const float* pc  = (const float*)d_in[0];
    float*       ws  = (float*)d_ws;     // 192 floats: [96 mins][96 maxs]
    float*       out = (float*)d_out;

    voxel_init_kernel<<<1, 96, 0, stream>>>(ws);
    voxel_minmax_kernel<<<dim3(B * R_BLOCKS_PER_BATCH), 256, 0, stream>>>(pc, ws);
    voxel_assign_kernel<<<dim3(B * A_BLOCKS_PER_BATCH), 256, 0, stream>>>(pc, ws, out);
}